// CPA_Loss_53996328846078
// MI455X (gfx1250) — compile-verified
//
#include <hip/hip_runtime.h>
#include <hip/hip_bf16.h>
#include <stdint.h>

// Problem constants (match reference)
#define B_ROWS   131072
#define C_CLS    128
#define D_DIM    64
#define NBLOCKS  512
#define ROWS_PER_WAVE 32   // 512 blocks * 8 waves * 32 rows = 131072

typedef __attribute__((ext_vector_type(2))) float v2f;
typedef __attribute__((ext_vector_type(8))) float v8f;

// ---- CDNA5 async global->LDS copy (ASYNCcnt-tracked) -----------------------
// GVS mode: saddr = 64-bit SGPR base, vaddr = 32-bit byte offset,
// vdst = LDS byte address (relative to wave LDS base). 16B per lane.
__device__ __forceinline__ void async_load_b128(uint32_t lds_addr, uint32_t goff,
                                                const void* sbase) {
  asm volatile("global_load_async_to_lds_b128 %0, %1, %2"
               :: "v"(lds_addr), "v"(goff), "s"((uint64_t)(uintptr_t)sbase)
               : "memory");
}
__device__ __forceinline__ void wait_async_le1() {
  asm volatile("s_wait_asynccnt 0x1" ::: "memory");
}
__device__ __forceinline__ void wait_async_le0() {
  asm volatile("s_wait_asynccnt 0x0" ::: "memory");
}

__global__ __launch_bounds__(256) void cpa_main(
    const float* __restrict__ logits,   // [B, C]
    const int*   __restrict__ targets,  // [B]
    const float* __restrict__ lproto,   // [C, D]
    const float* __restrict__ gproto,   // [C, D]
    const float* __restrict__ gfac,     // [C, C]
    float*       __restrict__ partials) // [NBLOCKS]
{
  __shared__ float stage[8 * 2 * C_CLS];  // per-wave double buffer, one row each
  __shared__ float pf_s[C_CLS];
  __shared__ float wacc[8];

  const int tid  = threadIdx.x;
  const int wave = tid >> 5;
  const int lane = tid & 31;

  const int row0 = blockIdx.x * (8 * ROWS_PER_WAVE) + wave * ROWS_PER_WAVE;
  float* mybuf = &stage[wave * 2 * C_CLS];
  const uint32_t lds0 = (uint32_t)(uintptr_t)mybuf;

  // Kick off the first row's async copy; it overlaps the WMMA phase below.
  async_load_b128(lds0 + lane * 16u,
                  (uint32_t)row0 * (C_CLS * 4u) + lane * 16u, logits);

  // ---- proto_factor via V_WMMA_F32_16X16X4_F32 row-sum trick (A @ ones) ----
  // Wave w computes rows [16w, 16w+16). A layout (f32 16x4): lanes 0-15 hold
  // (K0,K1) of row M=lane, lanes 16-31 hold (K2,K3) of row M=lane-16.
  {
    const int m    = wave * 16 + (lane & 15);
    const int koff = (lane < 16) ? 0 : 2;
    v8f cab = {}, caa = {}, cbb = {};
    v2f ones; ones.x = 1.0f; ones.y = 1.0f;
#pragma unroll
    for (int i = 0; i < 16; ++i) {   // K = 64 in steps of 4
      const int k = 4 * i + koff;
      v2f a = *(const v2f*)(lproto + m * D_DIM + k);
      v2f b = *(const v2f*)(gproto + m * D_DIM + k);
      v2f ab; ab.x = a.x * b.x; ab.y = a.y * b.y;
      v2f aa; aa.x = a.x * a.x; aa.y = a.y * a.y;
      v2f bb; bb.x = b.x * b.x; bb.y = b.y * b.y;
      // D[m,n] = rowsum(A) for every n; accumulate over K chunks via C.
      cab = __builtin_amdgcn_wmma_f32_16x16x4_f32(false, ab, false, ones, (short)0, cab, false, false);
      caa = __builtin_amdgcn_wmma_f32_16x16x4_f32(false, aa, false, ones, (short)0, caa, false, false);
      cbb = __builtin_amdgcn_wmma_f32_16x16x4_f32(false, bb, false, ones, (short)0, cbb, false, false);
    }
    // C/D layout: lane<16, VGPR r -> M=r ; lane>=16, VGPR r -> M=r+8.
    if (lane == 0 || lane == 16) {
      const int mb = wave * 16 + (lane == 16 ? 8 : 0);
#pragma unroll
      for (int r = 0; r < 8; ++r) {
        const float cosv = cab[r] / (sqrtf(caa[r]) * sqrtf(cbb[r]) + 1e-6f);
        pf_s[mb + r] = (1.0f + 3.0f) / (cosv + 3.0f);   // TAU = 3.0
      }
    }
  }
  __syncthreads();

  // ---- streaming phase: one row per wave-iteration, double-buffered --------
  float acc = 0.0f;
#pragma unroll 1
  for (int r = 0; r < ROWS_PER_WAVE; ++r) {
    const int row  = row0 + r;
    const int slot = r & 1;
    if (r + 1 < ROWS_PER_WAVE) {
      async_load_b128(lds0 + (uint32_t)((r + 1) & 1) * (C_CLS * 4u) + lane * 16u,
                      (uint32_t)(row + 1) * (C_CLS * 4u) + lane * 16u, logits);
      wait_async_le1();   // async loads complete in order: current row is ready
    } else {
      wait_async_le0();
    }

    const float4 xv = *(const float4*)(mybuf + slot * C_CLS + lane * 4);
    float mx = fmaxf(fmaxf(xv.x, xv.y), fmaxf(xv.z, xv.w));
#pragma unroll
    for (int o = 16; o >= 1; o >>= 1) mx = fmaxf(mx, __shfl_xor(mx, o, 32));

    const float e0 = __expf(xv.x - mx), e1 = __expf(xv.y - mx),
                e2 = __expf(xv.z - mx), e3 = __expf(xv.w - mx);

    const int tb = targets[row];                       // wave-uniform
    const float4 g = *(const float4*)(gfac + tb * C_CLS + lane * 4);  // L2-resident
    float ds = e0 * g.x + e1 * g.y + e2 * g.z + e3 * g.w;
#pragma unroll
    for (int o = 16; o >= 1; o >>= 1) ds += __shfl_xor(ds, o, 32);
    // ds == den[b, tb] (GF diagonal is 1.0, so the +e_tb term is included)

    const int c = tb & 3;
    const float esel = (c == 0) ? e0 : (c == 1) ? e1 : (c == 2) ? e2 : e3;
    const float etb  = __shfl(esel, tb >> 2, 32);

    const float sigma = etb / (ds + 1e-6f);
    acc += -pf_s[tb] * __logf(sigma + 1e-6f);          // identical in all lanes
  }

  if (lane == 0) wacc[wave] = acc;
  __syncthreads();
  if (tid == 0) {
    float s = 0.0f;
#pragma unroll
    for (int w = 0; w < 8; ++w) s += wacc[w];          // fixed order: deterministic
    partials[blockIdx.x] = s;
  }
}

__global__ __launch_bounds__(256) void cpa_finalize(
    const float* __restrict__ partials, float* __restrict__ out)
{
  __shared__ float sh[256];
  const int t = threadIdx.x;
  sh[t] = partials[t] + partials[t + 256];
  __syncthreads();
  for (int s = 128; s >= 1; s >>= 1) {                 // fixed tree: deterministic
    if (t < s) sh[t] += sh[t + s];
    __syncthreads();
  }
  if (t == 0) out[0] = sh[0] * (1.0f / (float)B_ROWS);
}

extern "C" void kernel_launch(void* const* d_in, const int* in_sizes, int n_in,
                              void* d_out, int out_size, void* d_ws, size_t ws_size,
                              hipStream_t stream) {
  const float* logits  = (const float*)d_in[0];
  const int*   targets = (const int*)  d_in[1];
  const float* lproto  = (const float*)d_in[2];
  const float* gproto  = (const float*)d_in[3];
  const float* gfac    = (const float*)d_in[4];
  float* out      = (float*)d_out;
  float* partials = (float*)d_ws;     // NBLOCKS floats = 2 KB

  cpa_main<<<NBLOCKS, 256, 0, stream>>>(logits, targets, lproto, gproto, gfac, partials);
  cpa_finalize<<<1, 256, 0, stream>>>(partials, out);
}